// AttentionMemory_84799834293119
// MI455X (gfx1250) — compile-verified
//
#include <hip/hip_runtime.h>
#include <hip/hip_bf16.h>

// Problem constants (match reference)
#define BB 16
#define DD 1024
#define MM 8192
#define HH 8
#define KDIM 64
#define VDIM 64
#define LL 8192
#define NN (MM + LL)          // 16384
#define HK (HH * KDIM)        // 512

typedef float v2f __attribute__((ext_vector_type(2)));
typedef float v8f __attribute__((ext_vector_type(8)));

#define LDS_STRIDE 68          // tile row stride (floats); keeps float4 align, avoids bank conflicts
#define WAVES_PER_BLK 4

// -------- cooperative 16x64 f32 tile load: global -> LDS (one wave) --------
__device__ __forceinline__ void load_tile16x64(float* dstLds, const float* src, int rowStride) {
    int lane  = threadIdx.x & 31;
    int row   = lane >> 1;           // 0..15
    int halfc = (lane & 1) * 32;     // 0 or 32
    const float4* s = (const float4*)(src + (size_t)row * rowStride + halfc);
    float4*       d = (float4*)(dstLds + row * LDS_STRIDE + halfc);
#pragma unroll
    for (int j = 0; j < 8; ++j) d[j] = s[j];
}

// Same, but also echoes the tile to a global destination (fused cache-append copy).
__device__ __forceinline__ void load_tile16x64_copy(float* dstLds, const float* src,
                                                    int rowStride, float* gdst, int dstRowStride) {
    int lane  = threadIdx.x & 31;
    int row   = lane >> 1;
    int halfc = (lane & 1) * 32;
    const float4* s = (const float4*)(src + (size_t)row * rowStride + halfc);
    float4*       d = (float4*)(dstLds + row * LDS_STRIDE + halfc);
    float4*       g = (float4*)(gdst + (size_t)row * dstRowStride + halfc);
#pragma unroll
    for (int j = 0; j < 8; ++j) { float4 v = s[j]; d[j] = v; g[j] = v; }
}

// -------- 16x16 score tile via V_WMMA_F32_16X16X4_F32, K accumulated to 64 --------
// A (16x4 f32): lanes 0-15 hold M=0..15 with VGPR0=K, VGPR1=K+1; lanes 16-31 K+2,K+3.
// B (4x16 f32): mirrored layout (lanes = N columns).
__device__ __forceinline__ v8f score_tile(const float* qlds, const float* klds) {
    v8f acc = {0.f, 0.f, 0.f, 0.f, 0.f, 0.f, 0.f, 0.f};
    int lane = threadIdx.x & 31;
    int kofs = (lane >> 4) * 2;      // 0 for lanes 0-15, 2 for lanes 16-31
    int r    = lane & 15;
    const float* qr = qlds + r * LDS_STRIDE;
    const float* kr = klds + r * LDS_STRIDE;
#pragma unroll
    for (int k = 0; k < KDIM; k += 4) {
        v2f a; a.x = qr[k + kofs]; a.y = qr[k + kofs + 1];
        v2f b; b.x = kr[k + kofs]; b.y = kr[k + kofs + 1];
        acc = __builtin_amdgcn_wmma_f32_16x16x4_f32(false, a, false, b, (short)0, acc,
                                                    false, false);
    }
    return acc;
}

// -------- QKV projections: q/nk/nv = x @ W + b  (3 x [16,512]) --------
__global__ void qkv_kernel(const float* __restrict__ x,
                           const float* __restrict__ Wq, const float* __restrict__ bq,
                           const float* __restrict__ Wk, const float* __restrict__ bk,
                           const float* __restrict__ Wv, const float* __restrict__ bv,
                           float* __restrict__ q, float* __restrict__ nk, float* __restrict__ nv) {
    int idx = blockIdx.x * blockDim.x + threadIdx.x;   // 3*16*512
    if (idx >= 3 * BB * HK) return;
    int which = idx / (BB * HK);
    int r     = idx - which * (BB * HK);
    int b     = r >> 9;
    int c     = r & (HK - 1);
    const float* W;  const float* bias;  float* dst;
    if (which == 0)      { W = Wq; bias = bq; dst = q;  }
    else if (which == 1) { W = Wk; bias = bk; dst = nk; }
    else                 { W = Wv; bias = bv; dst = nv; }
    float acc = bias[c];
    const float* xr = x + b * DD;
#pragma unroll 4
    for (int i = 0; i < DD; ++i) acc = fmaf(xr[i], W[i * HK + c], acc);
    dst[r] = acc;
}

__global__ void zero_kernel(float* __restrict__ p, int n) {
    int i = blockIdx.x * blockDim.x + threadIdx.x;
    if (i < n) p[i] = 0.f;
}

// -------- scores vs memory bank (shared across batch): WMMA GEMM --------
__global__ void score_mem_kernel(const float* __restrict__ q,
                                 const float* __restrict__ memK,
                                 float* __restrict__ scores) {
    __shared__ float lds[WAVES_PER_BLK][2][16 * LDS_STRIDE];
    int wave = threadIdx.x >> 5;
    int gw   = blockIdx.x * WAVES_PER_BLK + wave;      // H * (M/16) waves
    int h    = gw / (MM / 16);
    int tile = gw - h * (MM / 16);
    int n0   = tile * 16;

    load_tile16x64(lds[wave][0], q + h * KDIM, HK);                       // rows = batch
    load_tile16x64(lds[wave][1], memK + (size_t)n0 * HK + h * KDIM, HK);  // rows = key
#if __has_builtin(__builtin_amdgcn_wave_barrier)
    __builtin_amdgcn_wave_barrier();
#endif
    v8f acc = score_tile(lds[wave][0], lds[wave][1]);

    const float scale = 0.125f;                        // 1/sqrt(64)
    int lane = threadIdx.x & 31;
    int c    = lane & 15;
    int bhi  = (lane >> 4) * 8;
    // one base pointer; per-batch stride H*N is a compile-time constant -> immediate offsets
    float* base = scores + ((size_t)(bhi * HH + h)) * NN + n0 + c;
#pragma unroll
    for (int r = 0; r < 8; ++r)
        base[(size_t)r * (HH * NN)] = acc[r] * scale;
}

// -------- scores vs per-batch cache: WMMA (row b extracted) + fused key copy --------
__global__ void score_prev_kernel(const float* __restrict__ q,
                                  const float* __restrict__ prevK,
                                  float* __restrict__ scores,
                                  float* __restrict__ updK) {
    __shared__ float lds[WAVES_PER_BLK][2][16 * LDS_STRIDE];
    int wave = threadIdx.x >> 5;
    int gw   = blockIdx.x * WAVES_PER_BLK + wave;      // B*H*(L/16) waves
    int bh   = gw / (LL / 16);
    int tile = gw - bh * (LL / 16);
    int b    = bh / HH;
    int h    = bh - b * HH;
    int n0   = tile * 16;

    load_tile16x64(lds[wave][0], q + h * KDIM, HK);
    // key tile: stage into LDS AND echo into updated_keys[b, n0..n0+15, h, :]
    const float* srcK = prevK + ((size_t)b * LL + n0) * HK + h * KDIM;
    float*       dstK = updK + ((size_t)b * (LL + 1) + n0) * HK + h * KDIM;
    load_tile16x64_copy(lds[wave][1], srcK, HK, dstK, HK);
#if __has_builtin(__builtin_amdgcn_wave_barrier)
    __builtin_amdgcn_wave_barrier();
#endif
    v8f acc = score_tile(lds[wave][0], lds[wave][1]);

    const float scale = 0.125f;
    int lane = threadIdx.x & 31;
    int c    = lane & 15;
    if ((lane >> 4) == (b >> 3)) {                     // half-wave holding row b
        float s = acc[b & 7];
        scores[((size_t)(b * HH + h)) * NN + MM + n0 + c] = s * scale;
    }
}

// -------- softmax over N per (b,h); weights stored in-place, 1/sum saved --------
__global__ void softmax_kernel(float* __restrict__ scores, float* __restrict__ recip) {
    int bh = blockIdx.x;                               // 0..127
    float* s = scores + (size_t)bh * NN;
    __shared__ float red[256];
    int t = threadIdx.x;

    float mx = -1e30f;
    for (int i = t; i < NN; i += 256) mx = fmaxf(mx, s[i]);
    red[t] = mx; __syncthreads();
    for (int off = 128; off > 0; off >>= 1) {
        if (t < off) red[t] = fmaxf(red[t], red[t + off]);
        __syncthreads();
    }
    mx = red[0]; __syncthreads();

    float sum = 0.f;
    for (int i = t; i < NN; i += 256) {
        float e = __expf(s[i] - mx);
        s[i] = e;
        sum += e;
    }
    red[t] = sum; __syncthreads();
    for (int off = 128; off > 0; off >>= 1) {
        if (t < off) red[t] += red[t + off];
        __syncthreads();
    }
    if (t == 0) recip[bh] = 1.0f / red[0];
}

// -------- retrieved += sum_n w[n]*V[n,h,:]; prev chunks also copy V -> updated_values --------
#define NCHUNK 1024
__global__ void wv_kernel(const float* __restrict__ weights,
                          const float* __restrict__ memV,
                          const float* __restrict__ prevV,
                          float* __restrict__ retr,
                          float* __restrict__ updV) {
    int id    = blockIdx.x;                            // B*H*(N/NCHUNK)
    int chunk = id & (NN / NCHUNK - 1);                // 16 chunks
    int bh    = id / (NN / NCHUNK);
    int b     = bh / HH;
    int h     = bh - b * HH;
    int d     = threadIdx.x;                           // 0..63
    size_t n0 = (size_t)chunk * NCHUNK;

    const float* w = weights + (size_t)bh * NN + n0;
    float acc = 0.f;
    if (n0 < MM) {
        const float* v = memV + n0 * HK + h * VDIM + d;
#pragma unroll 4
        for (int i = 0; i < NCHUNK; ++i) acc = fmaf(w[i], v[(size_t)i * HK], acc);
    } else {
        size_t l0 = n0 - MM;
        const float* v = prevV + ((size_t)b * LL + l0) * HK + h * VDIM + d;
        float*       g = updV + ((size_t)b * (LL + 1) + l0) * HK + h * VDIM + d;
#pragma unroll 4
        for (int i = 0; i < NCHUNK; ++i) {
            float vv = v[(size_t)i * HK];
            acc = fmaf(w[i], vv, acc);
            g[(size_t)i * HK] = vv;                    // fused values cache-append copy
        }
    }
    atomicAdd(&retr[bh * VDIM + d], acc);
}

// -------- output = (retrieved * 1/sum) @ Wo + bo --------
__global__ void proj_kernel(const float* __restrict__ retr, const float* __restrict__ recip,
                            const float* __restrict__ Wo, const float* __restrict__ bo,
                            float* __restrict__ out) {
    int idx = blockIdx.x * blockDim.x + threadIdx.x;   // 16*1024
    if (idx >= BB * DD) return;
    int b = idx >> 10;
    int j = idx & (DD - 1);
    float acc = bo[j];
#pragma unroll 4
    for (int i = 0; i < HK; ++i)
        acc = fmaf(retr[b * HK + i] * recip[b * HH + (i >> 6)], Wo[i * DD + j], acc);
    out[idx] = acc;
}

// -------- append new row L: updK[b,L,:] = nk[b], updV[b,L,:] = nv[b] --------
__global__ void newrow_kernel(const float* __restrict__ nk, const float* __restrict__ nv,
                              float* __restrict__ updK, float* __restrict__ updV) {
    int idx = blockIdx.x * blockDim.x + threadIdx.x;   // 2*16*512
    if (idx >= 2 * BB * HK) return;
    int which = idx / (BB * HK);
    int r     = idx - which * (BB * HK);
    int b     = r >> 9;
    int j     = r & (HK - 1);
    size_t o  = ((size_t)b * (LL + 1) + LL) * HK + j;
    if (which == 0) updK[o] = nk[r];
    else            updV[o] = nv[r];
}

extern "C" void kernel_launch(void* const* d_in, const int* in_sizes, int n_in,
                              void* d_out, int out_size, void* d_ws, size_t ws_size,
                              hipStream_t stream) {
    const float* x     = (const float*)d_in[0];
    const float* prevK = (const float*)d_in[1];
    const float* prevV = (const float*)d_in[2];
    const float* Wq    = (const float*)d_in[3];
    const float* bq    = (const float*)d_in[4];
    const float* Wk    = (const float*)d_in[5];
    const float* bk    = (const float*)d_in[6];
    const float* Wv    = (const float*)d_in[7];
    const float* bv    = (const float*)d_in[8];
    const float* Wo    = (const float*)d_in[9];
    const float* bo    = (const float*)d_in[10];
    const float* memK  = (const float*)d_in[11];
    const float* memV  = (const float*)d_in[12];
    float* out = (float*)d_out;

    // d_out regions
    float* updK = out + BB * DD;                       // [16, 8193, 8, 64]
    float* updV = updK + (size_t)BB * (LL + 1) * HK;   // [16, 8193, 8, 64]

    // workspace layout (floats)
    float* ws     = (float*)d_ws;
    float* q      = ws;                         // 16*512
    float* nk     = q + BB * HK;                // 16*512
    float* nv     = nk + BB * HK;               // 16*512
    float* retr   = nv + BB * HK;               // 16*8*64
    float* recip  = retr + BB * HH * VDIM;      // 128
    float* scores = recip + BB * HH;            // 16*8*16384 (8 MB)

    // 1) projections + zero retrieved accumulator
    qkv_kernel<<<(3 * BB * HK + 255) / 256, 256, 0, stream>>>(x, Wq, bq, Wk, bk, Wv, bv,
                                                              q, nk, nv);
    zero_kernel<<<(BB * HH * VDIM + 255) / 256, 256, 0, stream>>>(retr, BB * HH * VDIM);

    // 2) WMMA score kernels (prev kernel also streams keys into updated_keys)
    {
        int waves  = HH * (MM / 16);                       // 4096
        int blocks = waves / WAVES_PER_BLK;
        score_mem_kernel<<<blocks, 32 * WAVES_PER_BLK, 0, stream>>>(q, memK, scores);
    }
    {
        int waves  = BB * HH * (LL / 16);                  // 65536
        int blocks = waves / WAVES_PER_BLK;
        score_prev_kernel<<<blocks, 32 * WAVES_PER_BLK, 0, stream>>>(q, prevK, scores, updK);
    }

    // 3) softmax per (b,h)
    softmax_kernel<<<BB * HH, 256, 0, stream>>>(scores, recip);

    // 4) weighted sum of values (prev chunks also stream values into updated_values)
    wv_kernel<<<BB * HH * (NN / NCHUNK), VDIM, 0, stream>>>(scores, memV, prevV, retr, updV);

    // 5) output projection
    proj_kernel<<<(BB * DD + 255) / 256, 256, 0, stream>>>(retr, recip, Wo, bo, out);

    // 6) append new K/V row (row L) — the only remaining copy work
    newrow_kernel<<<(2 * BB * HK + 255) / 256, 256, 0, stream>>>(nk, nv, updK, updV);
}